// HouseholderLinear_3083786518822
// MI455X (gfx1250) — compile-verified
//
#include <hip/hip_runtime.h>
#include <hip/hip_bf16.h>

typedef __attribute__((ext_vector_type(2))) float v2f;
typedef __attribute__((ext_vector_type(8))) float v8f;

#define FEATURES 1024
#define BATCH    65536

// ---------------------------------------------------------------------------
// Prologue (1 block): normalize vectors, build compact-WY T (4x4), and emit
//   ws[0      .. 16384): vhatP  — V-hat in WMMA-B pair-interleaved layout,
//                         padded to N=16 with zeros:
//                         vhatP[(f>>1)*32 + j*2 + (f&1)] = vhat_j[f] (j<4) else 0
//   ws[16384  .. 20480): wtP    — (-T * Vhat^T) pair-interleaved:
//                         wtP[((i>>1)*1024 + c)*2 + (i&1)] = -(T Vhat^T)[i][c]
// ---------------------------------------------------------------------------
__global__ __launch_bounds__(256) void hh_prologue(const float* __restrict__ vecs,
                                                   float* __restrict__ ws)
{
    __shared__ float g[16];      // Gram matrix v_i . v_k
    __shared__ float TL[16];     // T (4x4, row major)
    __shared__ float invn[4];    // 1/||v_i||

    const int t = threadIdx.x;
    if (t < 16) g[t] = 0.f;
    __syncthreads();

    // 16 (i,k) pairs x 16 threads each: partial dot products -> LDS atomics
    const int p  = t >> 4;
    const int i0 = p >> 2, k0 = p & 3;
    float part = 0.f;
    for (int f = (t & 15); f < FEATURES; f += 16)
        part += vecs[i0 * FEATURES + f] * vecs[k0 * FEATURES + f];
    atomicAdd(&g[p], part);
    __syncthreads();

    if (t == 0) {
        float inr[4];
        for (int a = 0; a < 4; ++a) inr[a] = 1.f / sqrtf(g[a * 4 + a]);
        float gh[4][4];
        for (int a = 0; a < 4; ++a)
            for (int b = 0; b < 4; ++b)
                gh[a][b] = g[a * 4 + b] * inr[a] * inr[b];
        // Compact WY: H0..Hk = I - V T V^T ;  T built column by column:
        //   T[kk][kk]=2 ; T[0..kk-1][kk] = -2 * T_{kk} * (Vhat^T vhat_kk)
        float T[4][4] = {};
        T[0][0] = 2.f;
        for (int kk = 1; kk < 4; ++kk) {
            for (int a = 0; a < kk; ++a) {
                float z = 0.f;
                for (int j = a; j < kk; ++j) z += T[a][j] * gh[j][kk];
                T[a][kk] = -2.f * z;
            }
            T[kk][kk] = 2.f;
        }
        for (int a = 0; a < 4; ++a) invn[a] = inr[a];
        for (int a = 0; a < 16; ++a) TL[a] = T[a >> 2][a & 3];
    }
    __syncthreads();

    float* vhatP = ws;
    float* wtP   = ws + 16384;
    for (int f = t; f < FEATURES; f += 256) {
        float vh[4];
        for (int j = 0; j < 4; ++j) vh[j] = vecs[j * FEATURES + f] * invn[j];
        const int base = (f >> 1) * 32 + (f & 1);
        #pragma unroll
        for (int j = 0; j < 16; ++j)
            vhatP[base + j * 2] = (j < 4) ? vh[j] : 0.f;
        #pragma unroll
        for (int ii = 0; ii < 4; ++ii) {
            float s = 0.f;
            for (int j = 0; j < 4; ++j) s += TL[ii * 4 + j] * vh[j];
            wtP[(((ii >> 1) * FEATURES) + f) * 2 + (ii & 1)] = -s;
        }
    }
}

// ---------------------------------------------------------------------------
// Main kernel: one wave per 16-row tile. Phase 1: S = X_tile * Vhat via
// v_wmma_f32_16x16x4_f32 accumulation over K=1024. Phase 2: per 16-column
// chunk, D = S * (-T Vhat^T)_chunk + X_chunk, streamed straight to out.
// ---------------------------------------------------------------------------
__global__ __launch_bounds__(256) void hh_main(const float* __restrict__ x,
                                               const float* __restrict__ ws,
                                               float* __restrict__ out)
{
    extern __shared__ float sm[];
    float* vhatP = sm;            // 16384 floats
    float* wtP   = sm + 16384;    //  4096 floats
    float* Sbuf  = sm + 20480;    //  2048 floats (8 waves x 256)

    // Stage operand tables into LDS (contiguous 80KB copy, all threads).
    for (int idx = threadIdx.x; idx < 20480; idx += 256) sm[idx] = ws[idx];
    __syncthreads();

    const int lane = threadIdx.x & 31;
    const int wave = threadIdx.x >> 5;
    const int half = lane >> 4;     // 0: K/M pair {0,1}, 1: pair {2,3}
    const int lm   = lane & 15;
    const int row0 = (blockIdx.x * 8 + wave) * 16;

    // ---- Phase 1: S = X_tile (16x1024) * Vhat (1024x4, padded to 16) ----
    // A layout (16x4 f32): lanes 0-15 hold rows, VGPR0/1 = K {0,1} (half 0)
    // or K {2,3} (half 1) -> per-lane float2 load at xrow[k + 2*half].
    const float* xrow = x + (size_t)(row0 + lm) * FEATURES;
    v8f acc0 = {}; v8f acc1 = {};
    #pragma unroll 4
    for (int k = 0; k < FEATURES; k += 8) {
        v2f a0 = *(const v2f*)(xrow + k + 2 * half);
        v2f b0 = *(const v2f*)(vhatP + ((k >> 1) + half) * 32 + lm * 2);
        acc0 = __builtin_amdgcn_wmma_f32_16x16x4_f32(false, a0, false, b0,
                                                     (short)0, acc0, false, false);
        v2f a1 = *(const v2f*)(xrow + k + 4 + 2 * half);
        v2f b1 = *(const v2f*)(vhatP + (((k + 4) >> 1) + half) * 32 + lm * 2);
        acc1 = __builtin_amdgcn_wmma_f32_16x16x4_f32(false, a1, false, b1,
                                                     (short)0, acc1, false, false);
    }
    v8f S = acc0 + acc1;   // C layout: VGPR r -> (M=r | r+8), lane -> N

    // ---- Re-stripe S (C layout) into A layout via per-wave LDS bounce ----
    float* sb = Sbuf + wave * 256;
    #pragma unroll
    for (int r = 0; r < 8; ++r)
        sb[(r + 8 * half) * 16 + lm] = S[r];
    __builtin_amdgcn_wave_barrier();
    asm volatile("s_wait_dscnt 0" ::: "memory");   // DS in-order per wave
    v2f aU = *(const v2f*)(sb + lm * 16 + 2 * half);   // (M=lm, i=2*half..+1)

    // ---- Phase 2: out_tile = S * Wtneg + X_tile, 64 chunks of 16 cols ----
    const size_t cbase = (size_t)(row0 + 8 * half) * FEATURES + lm;
    const float* xc = x   + cbase;
    float*       oc = out + cbase;
    #pragma unroll 2
    for (int jc = 0; jc < 64; ++jc) {
        v2f b = *(const v2f*)(wtP + (half * FEATURES + jc * 16 + lm) * 2);
        v8f c;
        #pragma unroll
        for (int r = 0; r < 8; ++r) c[r] = xc[(size_t)r * FEATURES + jc * 16];
        v8f d = __builtin_amdgcn_wmma_f32_16x16x4_f32(false, aU, false, b,
                                                      (short)0, c, false, false);
        #pragma unroll
        for (int r = 0; r < 8; ++r) oc[(size_t)r * FEATURES + jc * 16] = d[r];
    }
}

extern "C" void kernel_launch(void* const* d_in, const int* in_sizes, int n_in,
                              void* d_out, int out_size, void* d_ws, size_t ws_size,
                              hipStream_t stream)
{
    const float* x    = (const float*)d_in[0];   // (65536, 1024) f32
    const float* vecs = (const float*)d_in[1];   // (4, 1024) f32
    float* out = (float*)d_out;
    float* ws  = (float*)d_ws;                   // needs 20480 floats (80 KB)

    hipLaunchKernelGGL(hh_prologue, dim3(1), dim3(256), 0, stream, vecs, ws);
    // 4096 tiles of 16 rows, 8 waves (tiles) per block -> 512 blocks.
    hipLaunchKernelGGL(hh_main, dim3(512), dim3(256), 22528 * sizeof(float),
                       stream, x, ws, out);
}